// EngramModule_7378753815202
// MI455X (gfx1250) — compile-verified
//
#include <hip/hip_runtime.h>
#include <hip/hip_bf16.h>
#include <math.h>

typedef __attribute__((ext_vector_type(2))) float v2f;
typedef __attribute__((ext_vector_type(8))) float v8f;

#define B_      8
#define T_      4096
#define H_      4
#define E_      64
#define HR_     262144      // 2^18
#define HRMASK  (HR_ - 1)
#define D_      512
#define G_      256         // D/2
#define WAVES   8
#define LSTR    (D_ + 2)    // padded LDS row stride: banks (2*lane+k)%64, conflict-free

// ---------------------------------------------------------------------------
// Kernel 1: fp32-hash n-gram gather.
// seq_mem[pos, e] = 0.25 * sum_{n in {2,3}, h} E[h, hash_fp32(ngram)][e]
// 64 lanes per position -> each embedding row gather is a coalesced 256B read.
// Hash replicated in fp32 exactly like JAX (products ~5e8 round in fp32):
// left-associated adds, truncating float->int32, power-of-two mask.
// ---------------------------------------------------------------------------
__global__ __launch_bounds__(256) void ngram_gather_kernel(
    const int* __restrict__ tok, const float* __restrict__ emb,
    const int* __restrict__ seeds, float* __restrict__ seq_mem)
{
    int tid = threadIdx.x;
    int e   = tid & 63;
    int p   = tid >> 6;
    int pos = blockIdx.x * 4 + p;          // 0 .. B*T-1
    int b   = pos >> 12;                   // / T_
    int t   = pos & (T_ - 1);

    const int* trow = tok + (size_t)b * T_;
    int t0 = trow[t];
    int t1 = (t + 1 < T_) ? trow[t + 1] : 0;
    int t2 = (t + 2 < T_) ? trow[t + 2] : 0;

    float acc = 0.f;
#pragma unroll
    for (int h = 0; h < H_; ++h) {
        float sh = (float)(seeds[h] + 1);
        float s0 = (float)t0 * sh;         // fp32 product, rounds like JAX
        float s1 = (float)t1 * sh;
        float s2 = (float)t2 * sh;
        if (t <= T_ - 2) {
            float w2 = s0 + s1;            // left-associated like reference
            int idx2 = ((int)w2) & HRMASK; // trunc-toward-zero, power-of-2 mod
            acc += emb[((size_t)h * HR_ + idx2) * E_ + e];
            if (t <= T_ - 3) {
                float w3 = w2 + s2;
                int idx3 = ((int)w3) & HRMASK;
                acc += emb[((size_t)h * HR_ + idx3) * E_ + e];
            }
        }
    }
    seq_mem[(size_t)pos * E_ + e] = acc * 0.25f;   // mean over H=4 heads
}

// ---------------------------------------------------------------------------
// Kernel 2: fused fp32 WMMA GEMM chain + gate, one wave per 16-token tile.
// Both GEMMs: two N-tiles share each A-fragment; even/odd K-split gives
// 4 independent WMMA accumulator chains (fills the matrix pipe).
// ---------------------------------------------------------------------------
__global__ __launch_bounds__(WAVES * 32) void fused_gemm_gate_kernel(
    const float* __restrict__ hidden, const float* __restrict__ seq_mem,
    const float* __restrict__ W_hid,  const float* __restrict__ b_hid,
    const float* __restrict__ W_g1,   const float* __restrict__ b_g1,
    const float* __restrict__ W_g2,   const float* __restrict__ b_g2,
    float* __restrict__ out)
{
    __shared__ float g_lds[WAVES * 16 * LSTR];   // ~257 KB (<= 320 KB WGP LDS)
    __shared__ float gate_lds[WAVES * 16];

    int tid  = threadIdx.x;
    int wave = tid >> 5;
    int lane = tid & 31;
    int l15  = lane & 15;
    int hi   = lane >> 4;                      // 0/1: K pair select / row half

    size_t tile = (size_t)blockIdx.x * WAVES + wave;   // 0 .. B*T/16 - 1
    size_t row0 = tile * 16;                           // global token base

    float* gw = g_lds + wave * 16 * LSTR;

    // ---- GEMM1: A fragments (whole 16x64 seq_mem tile) stay in VGPRs ------
    v2f a1[E_ / 4];
#pragma unroll
    for (int k = 0; k < E_ / 4; ++k)
        a1[k] = *(const v2f*)(seq_mem + (row0 + l15) * E_ + k * 4 + hi * 2);

    for (int n0 = 0; n0 < D_; n0 += 32) {
        v8f acc0e = {}, acc0o = {}, acc1e = {}, acc1o = {};
        const float* w0 = W_hid + (size_t)(n0 + l15) * E_ + hi * 2;
        const float* w1 = w0 + 16 * E_;
#pragma unroll
        for (int k = 0; k < E_ / 4; k += 2) {
            v2f b0e = *(const v2f*)(w0 + k * 4);
            v2f b1e = *(const v2f*)(w1 + k * 4);
            v2f b0o = *(const v2f*)(w0 + k * 4 + 4);
            v2f b1o = *(const v2f*)(w1 + k * 4 + 4);
            acc0e = __builtin_amdgcn_wmma_f32_16x16x4_f32(false, a1[k],     false, b0e, (short)0, acc0e, false, false);
            acc1e = __builtin_amdgcn_wmma_f32_16x16x4_f32(false, a1[k],     false, b1e, (short)0, acc1e, false, false);
            acc0o = __builtin_amdgcn_wmma_f32_16x16x4_f32(false, a1[k + 1], false, b0o, (short)0, acc0o, false, false);
            acc1o = __builtin_amdgcn_wmma_f32_16x16x4_f32(false, a1[k + 1], false, b1o, (short)0, acc1o, false, false);
        }
        v8f acc0 = acc0e + acc0o;
        v8f acc1 = acc1e + acc1o;
        // C/D layout: VGPR r -> rows {r, r+8}, lane&15 -> column
        int d0    = n0 + l15;
        int d1    = d0 + 16;
        float bh0 = b_hid[d0];
        float bh1 = b_hid[d1];
#pragma unroll
        for (int r = 0; r < 8; ++r) {
            int row = r + hi * 8;
            float hv0 = hidden[(row0 + row) * D_ + d0];
            float hv1 = hidden[(row0 + row) * D_ + d1];
            gw[row * LSTR + d0] = hv0 + acc0[r] + bh0;   // g = hidden + mp + b
            gw[row * LSTR + d1] = hv1 + acc1[r] + bh1;
        }
    }

    // ---- GEMM2 + fused gelu + gate partial dot ----------------------------
    float gacc[8];
#pragma unroll
    for (int r = 0; r < 8; ++r) gacc[r] = 0.f;

    for (int n0 = 0; n0 < G_; n0 += 32) {
        v8f acc0e = {}, acc0o = {}, acc1e = {}, acc1o = {};
        const float* w0 = W_g1 + (size_t)(n0 + l15) * D_ + hi * 2;
        const float* w1 = w0 + 16 * D_;
        if (n0 + 32 < G_) {                    // prefetch next tile's rows
            __builtin_prefetch(w0 + 32 * D_, 0, 2);
            __builtin_prefetch(w1 + 32 * D_, 0, 2);
        }
#pragma unroll 2
        for (int k0 = 0; k0 < D_; k0 += 8) {
            v2f afe = *(const v2f*)(gw + l15 * LSTR + k0 + hi * 2);       // LDS
            v2f afo = *(const v2f*)(gw + l15 * LSTR + k0 + 4 + hi * 2);
            v2f b0e = *(const v2f*)(w0 + k0);
            v2f b1e = *(const v2f*)(w1 + k0);
            v2f b0o = *(const v2f*)(w0 + k0 + 4);
            v2f b1o = *(const v2f*)(w1 + k0 + 4);
            acc0e = __builtin_amdgcn_wmma_f32_16x16x4_f32(false, afe, false, b0e, (short)0, acc0e, false, false);
            acc1e = __builtin_amdgcn_wmma_f32_16x16x4_f32(false, afe, false, b1e, (short)0, acc1e, false, false);
            acc0o = __builtin_amdgcn_wmma_f32_16x16x4_f32(false, afo, false, b0o, (short)0, acc0o, false, false);
            acc1o = __builtin_amdgcn_wmma_f32_16x16x4_f32(false, afo, false, b1o, (short)0, acc1o, false, false);
        }
        v8f acc0 = acc0e + acc0o;
        v8f acc1 = acc1e + acc1o;
        int hc0   = n0 + l15;
        int hc1   = hc0 + 16;
        float bg0 = b_g1[hc0],  bg1 = b_g1[hc1];
        float wv0 = W_g2[hc0],  wv1 = W_g2[hc1];
#pragma unroll
        for (int r = 0; r < 8; ++r) {
            float x0  = acc0[r] + bg0;
            float x1  = acc1[r] + bg1;
            float ge0 = 0.5f * x0 * (1.f + erff(x0 * 0.70710678118654752f));
            float ge1 = 0.5f * x1 * (1.f + erff(x1 * 0.70710678118654752f));
            gacc[r] += ge0 * wv0 + ge1 * wv1;
        }
    }

    // reduce gate partials across the 16 lanes of each half-wave
#pragma unroll
    for (int r = 0; r < 8; ++r) {
        float v = gacc[r];
        v += __shfl_xor(v, 1);
        v += __shfl_xor(v, 2);
        v += __shfl_xor(v, 4);
        v += __shfl_xor(v, 8);
        gacc[r] = v;
    }
    float bg2 = b_g2[0];
    if (l15 == 0) {
#pragma unroll
        for (int r = 0; r < 8; ++r) {
            float z = gacc[r] + bg2;
            gate_lds[wave * 16 + hi * 8 + r] = 1.f / (1.f + expf(-z));
        }
    }

    // ---- Output: out = hidden + gate * (g - hidden) -----------------------
    for (int i = lane; i < 16 * D_; i += 32) {
        int row  = i >> 9;            // / D_
        int col  = i & (D_ - 1);
        float hv = hidden[(row0 + row) * D_ + col];
        float gv = gw[row * LSTR + col];
        float gt = gate_lds[wave * 16 + row];
        out[(row0 + row) * D_ + col] = hv + gt * (gv - hv);
    }
}

// ---------------------------------------------------------------------------
extern "C" void kernel_launch(void* const* d_in, const int* in_sizes, int n_in,
                              void* d_out, int out_size, void* d_ws, size_t ws_size,
                              hipStream_t stream)
{
    const int*   tok    = (const int*)d_in[0];
    const float* hidden = (const float*)d_in[1];
    const float* emb    = (const float*)d_in[2];
    const float* W_hid  = (const float*)d_in[3];
    const float* b_hid  = (const float*)d_in[4];
    const float* W_g1   = (const float*)d_in[5];
    const float* b_g1   = (const float*)d_in[6];
    const float* W_g2   = (const float*)d_in[7];
    const float* b_g2   = (const float*)d_in[8];
    const int*   seeds  = (const int*)d_in[9];
    // d_in[10] = hash_range (262144), d_in[11] = max_n (3): compile-time here.

    float* out     = (float*)d_out;
    float* seq_mem = (float*)d_ws;             // 8 MB scratch

    // Kernel 1: 4 positions per 256-thread block
    ngram_gather_kernel<<<(B_ * T_) / 4, 256, 0, stream>>>(tok, emb, seeds, seq_mem);

    // Kernel 2: 8 waves/block, one 16-token tile per wave
    fused_gemm_gate_kernel<<<(B_ * T_ / 16) / WAVES, WAVES * 32, 0, stream>>>(
        hidden, seq_mem, W_hid, b_hid, W_g1, b_g1, W_g2, b_g2, out);
}